// MinDistLoss_47519518163565
// MI455X (gfx1250) — compile-verified
//
#include <hip/hip_runtime.h>
#include <math.h>

typedef __attribute__((ext_vector_type(2))) float v2f;
typedef __attribute__((ext_vector_type(8))) float v8f;

#define BATCH 16
#define FLT_MAX_BITS 0x7F7FFFFFu

__global__ void mindist_init_kernel(unsigned int* ws) {
    if (threadIdx.x == 0 && blockIdx.x == 0) ws[0] = FLT_MAX_BITS;  // +FLT_MAX
}

// One wave per (batch, 32-row strip). Two A operands (2 row tiles), sweep all
// column tiles; per tile ONE b-vector load feeds TWO WMMAs:
//   A_n = (-2x0,-2x1,-2x2, ||x||^2),  B_m = (y0,y1,y2, 1),  C = 0
//   D_nm = ||x||^2 - 2 x.y          sq_nm = D_nm + ||y_m||^2
// Each lane of the 16x16 f32 D layout owns one column m, so per-lane:
//   min_m(sq) = min8(D) + yy[m]   (yy constant per lane -> add after min)
__global__ __launch_bounds__(256) void mindist_wmma_kernel(
    const float* __restrict__ v1, const float* __restrict__ v2,
    unsigned int* __restrict__ ws,
    int N, int M, int numRowPairs, int numColTilesPadded)
{
    const int lane = threadIdx.x & 31;
    const int wid  = (blockIdx.x * blockDim.x + threadIdx.x) >> 5;
    if (wid >= BATCH * numRowPairs) return;          // wave-uniform guard

    const int b  = wid / numRowPairs;
    const int rp = wid - b * numRowPairs;
    const bool lo = (lane < 16);                     // K=0,1 half vs K=2,3 half

    // ---- two A operands: rows rp*32 + lane%16 (+16), clamped (dups harmless)
    int row0 = rp * 32 + (lane & 15);
    int row1 = row0 + 16;
    if (row0 > N - 1) row0 = N - 1;
    if (row1 > N - 1) row1 = N - 1;

    const float* xp0 = v1 + ((size_t)b * N + row0) * 3;
    const float* xp1 = v1 + ((size_t)b * N + row1) * 3;

    float x00 = xp0[0], x01 = xp0[1], x02 = xp0[2];
    float x10 = xp1[0], x11 = xp1[1], x12 = xp1[2];
    float xx0 = x00 * x00 + x01 * x01 + x02 * x02;
    float xx1 = x10 * x10 + x11 * x11 + x12 * x12;

    v2f a0, a1;                         // -2 folded into A
    a0.x = lo ? (-2.0f * x00) : (-2.0f * x02);
    a0.y = lo ? (-2.0f * x01) : xx0;
    a1.x = lo ? (-2.0f * x10) : (-2.0f * x12);
    a1.y = lo ? (-2.0f * x11) : xx1;

    const v8f zeroc = {0.f, 0.f, 0.f, 0.f, 0.f, 0.f, 0.f, 0.f};
    const float* v2b = v2 + (size_t)b * M * 3;
    float mmin = 3.402823466e+38f;

    #pragma unroll 2
    for (int ct = 0; ct < numColTilesPadded; ++ct) {
        int col = ct * 16 + (lane & 15);
        if (col > M - 1) col = M - 1;
        const float* yp = v2b + (size_t)col * 3;
        const float y0 = yp[0], y1 = yp[1], y2 = yp[2];
        const float yy = y0 * y0 + y1 * y1 + y2 * y2;

        v2f bb;                          // B_m = (y0,y1,y2,1)
        bb.x = lo ? y0 : y2;
        bb.y = lo ? y1 : 1.0f;

        v8f d0 = __builtin_amdgcn_wmma_f32_16x16x4_f32(
            false, a0, false, bb, (short)0, zeroc, false, false);
        v8f d1 = __builtin_amdgcn_wmma_f32_16x16x4_f32(
            false, a1, false, bb, (short)0, zeroc, false, false);

        float t0 = fminf(fminf(fminf(d0[0], d0[1]), fminf(d0[2], d0[3])),
                         fminf(fminf(d0[4], d0[5]), fminf(d0[6], d0[7])));
        float t1 = fminf(fminf(fminf(d1[0], d1[1]), fminf(d1[2], d1[3])),
                         fminf(fminf(d1[4], d1[5]), fminf(d1[6], d1[7])));
        mmin = fminf(mmin, fminf(t0, t1) + yy);   // restore +||y||^2 per column
    }

    // wave32 butterfly min-reduce
    #pragma unroll
    for (int off = 16; off >= 1; off >>= 1)
        mmin = fminf(mmin, __shfl_xor(mmin, off, 32));

    if (lane == 0) {
        // clamp to >= 0 first: non-negative float bit patterns order as u32
        unsigned int bits = __float_as_uint(fmaxf(mmin, 0.0f));
        atomicMin(ws, bits);
    }
}

__global__ void mindist_final_kernel(const unsigned int* __restrict__ ws,
                                     float* __restrict__ out) {
    if (threadIdx.x == 0 && blockIdx.x == 0)
        out[0] = sqrtf(__uint_as_float(ws[0]));
}

extern "C" void kernel_launch(void* const* d_in, const int* in_sizes, int n_in,
                              void* d_out, int out_size, void* d_ws, size_t ws_size,
                              hipStream_t stream) {
    const float* v1 = (const float*)d_in[0];
    const float* v2 = (const float*)d_in[1];
    float* out = (float*)d_out;
    unsigned int* ws = (unsigned int*)d_ws;

    const int N = in_sizes[0] / (BATCH * 3);   // 6890
    const int M = in_sizes[1] / (BATCH * 3);   // 6890
    const int numRowPairs = (N + 31) / 32;     // 216 (two 16-row tiles per wave)
    const int numColTiles = (M + 15) / 16;     // 431
    const int numColTilesPadded = ((numColTiles + 1) / 2) * 2;  // 432 (unroll-2)

    const int totalWaves = BATCH * numRowPairs;               // 3456
    const int threads = 256;                                  // 8 waves / block
    const int blocks = (totalWaves * 32 + threads - 1) / threads;  // 432

    mindist_init_kernel<<<1, 1, 0, stream>>>(ws);
    mindist_wmma_kernel<<<blocks, threads, 0, stream>>>(
        v1, v2, ws, N, M, numRowPairs, numColTilesPadded);
    mindist_final_kernel<<<1, 1, 0, stream>>>(ws, out);
}